// ProgressiveRNN_45913200395013
// MI455X (gfx1250) — compile-verified
//
#include <hip/hip_runtime.h>
#include <hip/hip_bf16.h>
#include <math.h>

typedef __attribute__((ext_vector_type(2))) float v2f;
typedef __attribute__((ext_vector_type(8))) float v8f;

#define N_NODES 2048
#define BATCH   128
#define D_IN    2048
#define D_OUT   512

// NONLINEARITY_MAP: 0 ReLU, 1 Tanh, 2 LeakyReLU, 3 Identity, 4 Sine, 5 Softplus, 6 SoftMinus
__device__ __forceinline__ float act_f(int idx, float v) {
    switch (idx) {
        case 0:  return v > 0.f ? v : 0.f;
        case 1:  return tanhf(v);
        case 2:  return v > 0.f ? v : 0.01f * v;
        case 3:  return v;
        case 4:  return sinf(v);
        case 5:  return log1pf(expf(-fabsf(v))) + fmaxf(v, 0.f);           // softplus, overflow-safe
        default: return -(log1pf(expf(-fabsf(v))) + fmaxf(-v, 0.f));       // -softplus(-v)
    }
}

// ---------------------------------------------------------------------------
// K1: the scalar DAG recurrence. Single workgroup, s[] lives in LDS.
// Blocked 16 x 128: phase A = dense parallel prefix mat-vec (all 1024 threads),
// phase B = one wave resolves 128 sequential nodes with shfl reductions.
// Also emits t[k] = s[k] * adj[k, N-1] (mask folded for the output GEMM).
// ---------------------------------------------------------------------------
__global__ __launch_bounds__(1024)
void chain_kernel(const unsigned char* __restrict__ adj,
                  const float* __restrict__ Wmid,
                  const float* __restrict__ bmid,
                  const int* __restrict__ nonlin,
                  float* __restrict__ s_out,
                  float* __restrict__ t_out)
{
    __shared__ float s[N_NODES];
    __shared__ float partial[128];
    const int tid = threadIdx.x;

    for (int i = tid; i < N_NODES; i += 1024) s[i] = 0.f;
    __syncthreads();

    for (int blk = 0; blk < 16; ++blk) {
        const int nb = blk * 128;
        if (tid < 128) partial[tid] = 0.f;
        __syncthreads();

        // Phase A: partial[node-nb] += sum over finalized prefix [0, nb)
        if (nb > 0) {
            const int node  = nb + (tid >> 3);   // 128 nodes x 8 threads
            const int chunk = tid & 7;
            const float*         wrow = Wmid + (size_t)node * N_NODES;
            const unsigned char* acol = adj + node;           // adj[src*N + node]
            float acc = 0.f;
            for (int src = chunk; src < nb; src += 8) {
                if (acol[(size_t)src * N_NODES]) acc += wrow[src] * s[src];
            }
            atomicAdd(&partial[tid >> 3], acc);               // ds_add_f32
        }
        __syncthreads();

        // Phase B: wave 0 runs the 128 sequential steps of this block
        if (tid < 32) {
            const int lane = tid;
            for (int j = 0; j < 128; ++j) {
                const int node = nb + j;
                if (node >= 1 && node <= N_NODES - 2) {
                    const float* wrow = Wmid + (size_t)node * N_NODES;
                    float acc = 0.f;
                    for (int k = lane; k < j; k += 32) {      // intra-block srcs
                        const int src = nb + k;
                        if (adj[(size_t)src * N_NODES + node]) acc += wrow[src] * s[src];
                    }
                    #pragma unroll
                    for (int off = 16; off > 0; off >>= 1)
                        acc += __shfl_down(acc, off, 32);
                    if (lane == 0) {
                        float v = partial[j] + acc + bmid[node];
                        s[node] = act_f(nonlin[node], v);
                    }
                }
                __builtin_amdgcn_wave_barrier();              // keep LDS RAW order
            }
        }
        __syncthreads();
    }

    const unsigned char* lastcol = adj + (N_NODES - 1);
    for (int i = tid; i < N_NODES; i += 1024) {
        s_out[i] = s[i];
        t_out[i] = lastcol[(size_t)i * N_NODES] ? s[i] : 0.f;
    }
}

// ---------------------------------------------------------------------------
// K2: the only place x matters: state[:, N-1] = adj[0,N-1] ? act0(x . W0[N-1]) : 0
// ---------------------------------------------------------------------------
__global__ __launch_bounds__(256)
void col_last_kernel(const float* __restrict__ x,
                     const unsigned char* __restrict__ adj,
                     const int* __restrict__ nonlin,
                     const float* __restrict__ W0,
                     const float* __restrict__ b0,
                     float* __restrict__ state_out)
{
    __shared__ float red[256];
    const int b = blockIdx.x, tid = threadIdx.x;
    const float* xrow = x  + (size_t)b * D_IN;
    const float* wrow = W0 + (size_t)(N_NODES - 1) * D_IN;
    float acc = 0.f;
    for (int i = tid; i < D_IN; i += 256) acc += xrow[i] * wrow[i];
    red[tid] = acc;
    __syncthreads();
    for (int sft = 128; sft > 0; sft >>= 1) {
        if (tid < sft) red[tid] += red[tid + sft];
        __syncthreads();
    }
    if (tid == 0) {
        float v = act_f(nonlin[0], red[0] + b0[N_NODES - 1]);
        state_out[(size_t)b * N_NODES + (N_NODES - 1)] = adj[N_NODES - 1] ? v : 0.f;
    }
}

// ---------------------------------------------------------------------------
// K3: broadcast s[] into state rows (columns 0..N-2; K2 owns column N-1)
// ---------------------------------------------------------------------------
__global__ __launch_bounds__(256)
void bcast_kernel(const float* __restrict__ s, float* __restrict__ state_out)
{
    const int i   = blockIdx.x * 256 + threadIdx.x;   // 0 .. BATCH*N_NODES
    const int col = i & (N_NODES - 1);
    if (col != N_NODES - 1) state_out[i] = s[col];
}

// ---------------------------------------------------------------------------
// K4: out = act(nl_last, state @ (Wlast*mask)^T + blast) as an f32 WMMA GEMM.
// Mask folded into broadcast A-vector t[]; B = raw Wlast, contiguous per row.
// One wave per 16x16 tile; K-loop 2048/4 v_wmma_f32_16x16x4_f32 issues.
// A frag (16x4 f32): lanes 0-15 hold K=0,1 ; lanes 16-31 hold K=2,3.
// B frag (4x16 f32): mirrored, N striped across lanes.
// C/D: VGPR r -> M = 8*half + r, N = lane&15 (+n0).
// ---------------------------------------------------------------------------
__global__ __launch_bounds__(32)
void out_gemm_kernel(const float* __restrict__ t,
                     const float* __restrict__ Wlast,
                     const float* __restrict__ blast,
                     const int* __restrict__ nonlin,
                     float* __restrict__ out)
{
    const int lane = threadIdx.x;
    const int half = lane >> 4;
    const int l    = lane & 15;
    const int m0   = (blockIdx.x & 7) * 16;    // 128/16 = 8 M-tiles
    const int n0   = (blockIdx.x >> 3) * 16;   // 512/16 = 32 N-tiles
    const float* brow = Wlast + (size_t)(n0 + l) * N_NODES;

    v8f c = {};
    for (int k0 = 0; k0 < N_NODES; k0 += 4) {
        const int ka = k0 + 2 * half;
        const float2 av = *(const float2*)(t + ka);     // A[m, ka..ka+1] (rows identical)
        const float2 bv = *(const float2*)(brow + ka);  // B[ka..ka+1, n0+l] = Wlast[n0+l, ka..]
        v2f a; a[0] = av.x; a[1] = av.y;
        v2f b; b[0] = bv.x; b[1] = bv.y;
        c = __builtin_amdgcn_wmma_f32_16x16x4_f32(false, a, false, b,
                                                  (short)0, c, false, false);
    }

    const int   nl = nonlin[N_NODES - 1];
    const int   n  = n0 + l;
    const float bb = blast[n];
    #pragma unroll
    for (int r = 0; r < 8; ++r) {
        const int m = m0 + half * 8 + r;
        out[(size_t)m * D_OUT + n] = act_f(nl, c[r] + bb);
    }
}

extern "C" void kernel_launch(void* const* d_in, const int* in_sizes, int n_in,
                              void* d_out, int out_size, void* d_ws, size_t ws_size,
                              hipStream_t stream) {
    const float*         x      = (const float*)d_in[0];
    const unsigned char* adj    = (const unsigned char*)d_in[1];  // bool -> 1 byte
    const int*           nonlin = (const int*)d_in[2];
    const float*         W0     = (const float*)d_in[3];
    const float*         b0     = (const float*)d_in[4];
    const float*         Wmid   = (const float*)d_in[5];
    const float*         bmid   = (const float*)d_in[6];
    const float*         Wlast  = (const float*)d_in[7];
    const float*         blast  = (const float*)d_in[8];

    float* out       = (float*)d_out;               // [128, 512]
    float* state_out = out + BATCH * D_OUT;         // [128, 2048]
    float* s         = (float*)d_ws;                // [2048]
    float* t         = s + N_NODES;                 // [2048] masked

    hipLaunchKernelGGL(chain_kernel, dim3(1), dim3(1024), 0, stream,
                       adj, Wmid, bmid, nonlin, s, t);
    hipLaunchKernelGGL(col_last_kernel, dim3(BATCH), dim3(256), 0, stream,
                       x, adj, nonlin, W0, b0, state_out);
    hipLaunchKernelGGL(bcast_kernel, dim3(BATCH * N_NODES / 256), dim3(256), 0, stream,
                       s, state_out);
    hipLaunchKernelGGL(out_gemm_kernel, dim3((BATCH / 16) * (D_OUT / 16)), dim3(32), 0, stream,
                       t, Wlast, blast, nonlin, out);
}